// ConvIntrinsic_81183471829214
// MI455X (gfx1250) — compile-verified
//
#include <hip/hip_runtime.h>

// CDNA5 / gfx1250. Wave32. Uses V_WMMA_F32_16X16X4_F32 (exact f32 matmul).

typedef float v2f __attribute__((ext_vector_type(2)));
typedef float v8f __attribute__((ext_vector_type(8)));

#define RDIM 5
#define ADIM 8
#define FDIM 16
#define TDIM 32
#define ODIM 8
#define NCOL 256          // O*T
#define KTOT 656          // R*A*F + F  (interp columns + self columns)
#define XSTRIDE 660       // padded LDS row stride (conflict-free for WMMA A frags)
#define PT_PER_BLK 4      // 4 row-tiles of 16 points => 64 points per block

// ---------------------------------------------------------------------------
// Kernel A: fold prior kernel + rotated neighbor weights (+ self weights)
// into M'[KTOT][NCOL], stored FRAGMENT-SWIZZLED for b64 B-fragment loads:
//   logical  row < 640 : row = (x*8+y)*16+f ; col = o*32+t
//       M'[row][col] = sum_{r,a} kernel[r,a,x,y] * nw[t, r, (a+o)%8, f]
//   logical  row >= 640: row = 640+f ; M'[row][col] = self_weights[t, 0, f]
// Swizzle: kk = row>>2 (k-step), h = (row>>1)&1 (lane half), w = row&1
//   Msw[((kk*2 + h)*NCOL + col)*2 + w] = M'[row][col]
// so a lane's (k, k+1) B pair is one contiguous 8-byte load.
// ---------------------------------------------------------------------------
__global__ __launch_bounds__(NCOL) void fold_M_kernel(
    const float* __restrict__ kern,   // (R,A,R,A)
    const float* __restrict__ nw,     // (T,R,A,F)
    const float* __restrict__ sw,     // (T,1,F)
    float* __restrict__ Msw)          // KTOT*NCOL floats, swizzled
{
    const int row = blockIdx.x;       // 0..655
    const int col = threadIdx.x;      // 0..255
    float acc;
    if (row < 640) {
        const int xy = row >> 4, f = row & 15;
        const int x = xy >> 3, y = xy & 7;
        const int o = col >> 5, t = col & 31;
        acc = 0.0f;
#pragma unroll
        for (int r = 0; r < RDIM; ++r) {
#pragma unroll
            for (int a = 0; a < ADIM; ++a) {
                const float kv = kern[((r * ADIM + a) * RDIM + x) * ADIM + y];
                const float wv = nw[((t * RDIM + r) * ADIM + ((a + o) & 7)) * FDIM + f];
                acc = fmaf(kv, wv, acc);
            }
        }
    } else {
        const int f = row - 640;
        const int t = col & 31;
        acc = sw[t * FDIM + f];
    }
    const int kk = row >> 2;
    const int h  = (row >> 1) & 1;
    const int w  = row & 1;
    Msw[(((kk * 2 + h) * NCOL) + col) * 2 + w] = acc;
}

// ---------------------------------------------------------------------------
// Kernel B: per block of 64 points:
//   Phase 1: gather + barycentric interp -> Xs[4][16][660] in LDS
//            (row n: 640 interp values + 16 raw mesh features)
//   Phase 2: 8 waves x (4 row-tiles x 2 col-tiles) WMMA f32 16x16x4 GEMM
//            against swizzled M' (L2 resident), bias + relu epilogue.
// ---------------------------------------------------------------------------
__global__ __launch_bounds__(256) void conv_main_kernel(
    const float* __restrict__ mesh,   // (N, F)
    const float* __restrict__ bary,   // (N, R, A, 3, 2)
    const float* __restrict__ bias,   // (T)
    const float* __restrict__ Msw,    // swizzled M' (KTOT x NCOL)
    float* __restrict__ out,          // (N, O, T)
    int N, int ntiles)
{
    __shared__ float Xs[PT_PER_BLK][16][XSTRIDE];

    const int tid   = threadIdx.x;
    const int tile0 = blockIdx.x * PT_PER_BLK;
    int nt = ntiles - tile0;
    if (nt > PT_PER_BLK) nt = PT_PER_BLK;
    const int n0 = tile0 * 16;

    // ---------------- Phase 1: build X rows ----------------
    {
        const int f = tid & 15;       // feature lane
        const int g = tid >> 4;       // 16 task groups
        const int ntasks = nt * 16 * 41;   // per point: 40 taps + 1 self group
        for (int task = g; task < ntasks; task += 16) {
            const int p   = task / 41;
            const int sub = task - p * 41;
            const int n   = n0 + p;
            if (n < N) {
                const int pt = p >> 4, pr = p & 15;
                if (sub < 40) {
                    const long base = ((long)n * 40 + sub) * 6;
                    const float i0f = bary[base + 0], w0 = bary[base + 1];
                    const float i1f = bary[base + 2], w1 = bary[base + 3];
                    const float i2f = bary[base + 4], w2 = bary[base + 5];
                    const int i0 = (int)i0f, i1 = (int)i1f, i2 = (int)i2f;
                    float v = w0 * mesh[i0 * FDIM + f];
                    v = fmaf(w1, mesh[i1 * FDIM + f], v);
                    v = fmaf(w2, mesh[i2 * FDIM + f], v);
                    Xs[pt][pr][sub * FDIM + f] = v;
                } else {
                    Xs[pt][pr][640 + f] = mesh[(long)n * FDIM + f];
                }
            }
        }
    }
    __syncthreads();

    // ---------------- Phase 2: WMMA GEMM ----------------
    const int wave  = tid >> 5;            // 0..7 -> 32-column strip
    const int lane  = tid & 31;
    const int lanem = lane & 15;
    const int half  = lane >> 4;           // 0 or 1
    const int khalf = half << 1;           // A/B fragment K offset: 0 or 2
    const int col0  = wave * 32;

    const v2f* __restrict__ Bsw = (const v2f*)Msw;

    v8f acc[PT_PER_BLK][2];
#pragma unroll
    for (int pt = 0; pt < PT_PER_BLK; ++pt) {
        acc[pt][0] = (v8f)(0.0f);
        acc[pt][1] = (v8f)(0.0f);
    }

    for (int k0 = 0; k0 < KTOT; k0 += 4) {
        const int kA = k0 + khalf;
        const int kk = k0 >> 2;
        // B fragments: one contiguous v2f (b64) load per fragment
        const long bidx = (long)(kk * 2 + half) * NCOL + col0 + lanem;
        v2f b0 = Bsw[bidx];
        v2f b1 = Bsw[bidx + 16];
#pragma unroll
        for (int pt = 0; pt < PT_PER_BLK; ++pt) {
            v2f a;
            a.x = Xs[pt][lanem][kA];
            a.y = Xs[pt][lanem][kA + 1];
            acc[pt][0] = __builtin_amdgcn_wmma_f32_16x16x4_f32(
                false, a, false, b0, (short)0, acc[pt][0], false, false);
            acc[pt][1] = __builtin_amdgcn_wmma_f32_16x16x4_f32(
                false, a, false, b1, (short)0, acc[pt][1], false, false);
        }
    }

    // ---------------- Epilogue: bias + relu + store ----------------
    // C/D layout: VGPR v -> row = v + 8*(lane>=16), col = col0 + (lane&15)
    const float b_lo = bias[lanem];        // t = (col0 + lanem) & 31 = lanem
    const float b_hi = bias[lanem + 16];
    const int rbase = half << 3;
    for (int pt = 0; pt < nt; ++pt) {
#pragma unroll
        for (int v = 0; v < 8; ++v) {
            const int n = n0 + pt * 16 + rbase + v;
            if (n < N) {
                const float x0 = fmaxf(acc[pt][0][v] + b_lo, 0.0f);
                const float x1 = fmaxf(acc[pt][1][v] + b_hi, 0.0f);
                const long o = (long)n * NCOL + col0 + lanem;
                out[o]      = x0;
                out[o + 16] = x1;
            }
        }
    }
}

extern "C" void kernel_launch(void* const* d_in, const int* in_sizes, int n_in,
                              void* d_out, int out_size, void* d_ws, size_t ws_size,
                              hipStream_t stream) {
    const float* mesh = (const float*)d_in[0];   // (B,N,F) f32
    const float* bary = (const float*)d_in[1];   // (B,N,R,A,3,2) f32
    const float* nw   = (const float*)d_in[2];   // (T,R,A,F) f32
    const float* sw   = (const float*)d_in[3];   // (T,1,F) f32
    const float* bias = (const float*)d_in[4];   // (T,) f32
    const float* kern = (const float*)d_in[5];   // (R,A,R,A) f32
    float* out   = (float*)d_out;                // (B,N,O,T) f32
    float* Mfold = (float*)d_ws;                 // KTOT*NCOL f32 = 656 KiB scratch

    const int N = in_sizes[0] / FDIM;            // number of mesh points
    const int ntiles = (N + 15) / 16;
    const int nblocks = (ntiles + PT_PER_BLK - 1) / PT_PER_BLK;

    fold_M_kernel<<<KTOT, NCOL, 0, stream>>>(kern, nw, sw, Mfold);
    conv_main_kernel<<<nblocks, 256, 0, stream>>>(mesh, bary, bias, Mfold, out, N, ntiles);
}